// TexualFusedEmbeddingLayer_21036749815973
// MI455X (gfx1250) — compile-verified
//
#include <hip/hip_runtime.h>
#include <hip/hip_bf16.h>
#include <math.h>

// Problem constants (match reference)
#define BB 128
#define LL 512
#define DD 512
#define EE 1024
#define HH 512
#define KK 153   // int((512-2)*0.3)
#define KP 160   // K padded to a multiple of 16 (10 WMMA row tiles)

typedef __attribute__((ext_vector_type(2))) float v2f;
typedef __attribute__((ext_vector_type(8))) float v8f;

typedef int v4i __attribute__((vector_size(16)));
typedef __attribute__((address_space(1))) v4i gas_v4i;
typedef __attribute__((address_space(3))) v4i las_v4i;
typedef __attribute__((address_space(3))) char las_char;

__device__ __forceinline__ v8f wmma_f32_4(v2f a, v2f b, v8f c) {
  // V_WMMA_F32_16X16X4_F32 : D = A(16x4) x B(4x16) + C(16x16), fp32 throughout
  return __builtin_amdgcn_wmma_f32_16x16x4_f32(
      /*neg_a=*/false, a, /*neg_b=*/false, b,
      /*c_mod=*/(short)0, c, /*reuse_a=*/false, /*reuse_b=*/false);
}

// Async global -> LDS copy of 16 bytes (GLOBAL_LOAD_ASYNC_TO_LDS_B128,
// tracked by ASYNCcnt). Builtin if available, inline asm otherwise.
__device__ __forceinline__ void async_ld16(const float* g, float* l) {
#if __has_builtin(__builtin_amdgcn_global_load_async_to_lds_b128)
  __builtin_amdgcn_global_load_async_to_lds_b128(
      (gas_v4i*)g, (las_v4i*)l, 0, 0);
#else
  unsigned loff = (unsigned)(unsigned long long)(las_char*)l;
  asm volatile("global_load_async_to_lds_b128 %0, %1, off"
               :: "v"(loff), "v"(g) : "memory");
#endif
}

__device__ __forceinline__ void wait_async0() {
#if __has_builtin(__builtin_amdgcn_s_wait_asynccnt)
  __builtin_amdgcn_s_wait_asynccnt(0);
#else
  asm volatile("s_wait_asynccnt 0x0" ::: "memory");
#endif
}

// ---------------------------------------------------------------------------
// Kernel 1: per-batch prep. argmax(text) -> eos, nonzero count -> lens,
// masked attention row -> full bitonic sort (desc value, asc index) in LDS,
// emit top-K indices, and copy the global feature row.
// One block (512 threads = L) per batch.
// ---------------------------------------------------------------------------
__global__ __launch_bounds__(512)
void k1_prep(const float* __restrict__ gfeat, const int* __restrict__ text,
             const float* __restrict__ atten, float* __restrict__ gf,
             int* __restrict__ idxbuf, int* __restrict__ lens,
             int* __restrict__ eosbuf) {
  __shared__ unsigned long long skey[LL];
  __shared__ unsigned long long smax;
  __shared__ int scnt;
  const int b = blockIdx.x;
  const int tid = threadIdx.x;
  if (tid == 0) { smax = 0ull; scnt = 0; }
  __syncthreads();

  const int tv = text[b * LL + tid];
  // argmax with first-occurrence tie-break: max over (val<<32 | (L-1-tid))
  unsigned long long akey =
      ((unsigned long long)(unsigned)tv << 32) | (unsigned)(LL - 1 - tid);
  atomicMax(&smax, akey);
  atomicAdd(&scnt, (tv != 0) ? 1 : 0);
  __syncthreads();

  const int eos = (LL - 1) - (int)(unsigned)(smax & 0xffffffffull);

  // row = atten[b, eos, :]; row[eos] = -1; row[0] = -1; row *= (text != 0)
  float val = atten[((size_t)b * LL + (size_t)eos) * LL + tid];
  if (tid == eos) val = -1.0f;
  if (tid == 0)   val = -1.0f;
  if (tv == 0)    val = 0.0f;

  // order-preserving float->uint, pack with (L-1-idx) so that a descending
  // u64 sort yields (value desc, index asc) exactly like jax.lax.top_k
  unsigned fb  = __float_as_uint(val);
  unsigned ord = (fb & 0x80000000u) ? ~fb : (fb | 0x80000000u);
  skey[tid] = ((unsigned long long)ord << 32) | (unsigned)(LL - 1 - tid);

  // bitonic sort, descending, 512 elements, 512 threads
  for (int k = 2; k <= LL; k <<= 1) {
    for (int j = k >> 1; j > 0; j >>= 1) {
      __syncthreads();
      int ixj = tid ^ j;
      if (ixj > tid) {
        unsigned long long x = skey[tid], y = skey[ixj];
        bool descBlk = ((tid & k) == 0);
        if (descBlk ? (y > x) : (x > y)) { skey[tid] = y; skey[ixj] = x; }
      }
    }
  }
  __syncthreads();

  if (tid < KK)
    idxbuf[b * KK + tid] = (LL - 1) - (int)(unsigned)(skey[tid] & 0xffffffffull);
  if (tid == 0) {
    int lg = scnt - 2;
    lg = lg < 0 ? 0 : lg;
    lens[b] = lg < KK ? lg : KK;
    eosbuf[b] = eos;
  }
  // global_features[b, :] = gfeatures[b, eos, :]
  gf[(size_t)b * DD + tid] = gfeat[((size_t)b * LL + (size_t)eos) * DD + tid];
}

// ---------------------------------------------------------------------------
// Kernel 2: gather + L2-normalize selected rows into padded A matrix
// (B*KP x D). Rows k in [K, KP) are zero-filled.
// One block (256 threads) per (b, k).
// ---------------------------------------------------------------------------
__global__ __launch_bounds__(256)
void k2_gather_norm(const float* __restrict__ gfeat,
                    const int* __restrict__ idxbuf,
                    float* __restrict__ localn) {
  __shared__ float red[256];
  const int bk = blockIdx.x;
  const int b = bk / KP;
  const int k = bk % KP;
  const int tid = threadIdx.x;
  float* dst = localn + ((size_t)b * KP + k) * DD;
  if (k >= KK) { dst[tid] = 0.0f; dst[tid + 256] = 0.0f; return; }
  const int row = idxbuf[b * KK + k];
  const float* src = gfeat + ((size_t)b * LL + (size_t)row) * DD;
  float x0 = src[tid], x1 = src[tid + 256];
  red[tid] = x0 * x0 + x1 * x1;
  __syncthreads();
  for (int s = 128; s > 0; s >>= 1) {
    if (tid < s) red[tid] += red[tid + s];
    __syncthreads();
  }
  float inv = 1.0f / (sqrtf(red[0]) + 1e-8f);
  dst[tid] = x0 * inv;
  dst[tid + 256] = x1 * inv;
}

// ---------------------------------------------------------------------------
// Kernel 3: GEMM1 (B*KP x D) @ (D x H) + bias + BatchNorm + ReLU -> hrelu.
// Block-cooperative 128x64 tile, double-buffered async global->LDS staging
// (GLOBAL_LOAD_ASYNC_TO_LDS_B128 + s_wait_asynccnt). 8 waves share the
// LDS-resident weight tile; each wave owns a 16-row sub-tile (4 accumulators).
// ---------------------------------------------------------------------------
#define BM 128
#define BN 64
#define KB 16
#define A_STR 20   // floats; 80 B row stride: 16B-aligned b128 stores, no bank dup
#define B_STR 72   // floats; 288 B row stride: 16B-aligned, half-waves on disjoint banks

__global__ __launch_bounds__(256)
void k3_gemm1(const float* __restrict__ A, const float* __restrict__ w1,
              const float* __restrict__ b1, const float* __restrict__ g1,
              const float* __restrict__ be1, const float* __restrict__ rm1,
              const float* __restrict__ rv1, float* __restrict__ hrelu) {
  __shared__ float As[2][BM * A_STR];
  __shared__ float Bs[2][KB * B_STR];

  const int tid  = threadIdx.x;
  const int wid  = tid >> 5;
  const int lane = tid & 31;
  const int nBase = blockIdx.x * BN;        // 0..511 step 64
  const int mBase = blockIdx.y * BM;        // 0..20479 step 128
  const int bn   = lane & 15;               // A row in sub-tile / B col in tile
  const int acol = (lane >> 4) << 1;        // lanes<16 -> K 0,1 ; else K 2,3
  const int brow = (lane >> 4) << 1;

  // per-thread async-copy coordinates (A: 512 x b128 chunks, B: 256 chunks)
  const int ar0 = tid >> 2;                 // A rows 0..63
  const int ar1 = ar0 + 64;                 // A rows 64..127
  const int ac  = (tid & 3) << 2;           // A col 0,4,8,12
  const int br  = tid >> 4;                 // B row 0..15
  const int bc  = (tid & 15) << 2;          // B col 0..60 step 4

  auto issue = [&](int buf, int k0) {
    async_ld16(A + (size_t)(mBase + ar0) * DD + k0 + ac, &As[buf][ar0 * A_STR + ac]);
    async_ld16(A + (size_t)(mBase + ar1) * DD + k0 + ac, &As[buf][ar1 * A_STR + ac]);
    async_ld16(w1 + (size_t)(k0 + br) * HH + nBase + bc, &Bs[buf][br * B_STR + bc]);
  };

  v8f acc0 = {}, acc1 = {}, acc2 = {}, acc3 = {};

  issue(0, 0);
  wait_async0();
  __syncthreads();

  const int NS = DD / KB;                   // 32 K-steps
  for (int ks = 0; ks < NS; ++ks) {
    const int cur = ks & 1;
    if (ks + 1 < NS) issue(cur ^ 1, (ks + 1) * KB);

    const float* as = &As[cur][(wid * 16 + bn) * A_STR];
    const float* bs = &Bs[cur][0];
#pragma unroll
    for (int kk = 0; kk < KB; kk += 4) {
      v2f a = *(const v2f*)(as + kk + acol);
      const float* bp = bs + (kk + brow) * B_STR + bn;
      v2f bf;
      bf.x = bp[0];       bf.y = bp[B_STR];
      acc0 = wmma_f32_4(a, bf, acc0);
      bf.x = bp[16];      bf.y = bp[B_STR + 16];
      acc1 = wmma_f32_4(a, bf, acc1);
      bf.x = bp[32];      bf.y = bp[B_STR + 32];
      acc2 = wmma_f32_4(a, bf, acc2);
      bf.x = bp[48];      bf.y = bp[B_STR + 48];
      acc3 = wmma_f32_4(a, bf, acc3);
    }
    wait_async0();
    __syncthreads();
  }

  // epilogue: bias + BN + ReLU, wave wid owns rows mBase + wid*16 .. +15
  const int m0 = mBase + wid * 16;
  const int rbase = (lane >> 4) << 3;       // C/D: lanes<16 rows 0-7, else 8-15
  v8f accs[4] = {acc0, acc1, acc2, acc3};
#pragma unroll
  for (int j = 0; j < 4; ++j) {
    const int col = nBase + j * 16 + bn;
    const float bb = b1[col];
    const float sc = g1[col] * rsqrtf(rv1[col] + 1e-5f);
    const float sh = be1[col] - rm1[col] * sc;
    v8f acc = accs[j];
#pragma unroll
    for (int r = 0; r < 8; ++r) {
      const int row = m0 + rbase + r;
      float v = acc[r] + bb;
      v = v * sc + sh;
      v = v > 0.0f ? v : 0.0f;
      hrelu[(size_t)row * HH + col] = v;
    }
  }
}

// ---------------------------------------------------------------------------
// Kernel 4: GEMM2 per batch fused with validity-masked max-pool.
// One wave per (batch, 16-column tile of E); loops 10 row tiles x K=512.
// max(h@w2 + b2) == max(h@w2) + b2, so bias is added after pooling.
// ---------------------------------------------------------------------------
__global__ __launch_bounds__(32)
void k4_gemm2_pool(const float* __restrict__ hrelu, const float* __restrict__ w2,
                   const float* __restrict__ b2, const int* __restrict__ lens,
                   float* __restrict__ pooled) {
  const int lane = threadIdx.x & 31;
  const int n0 = blockIdx.x * 16;
  const int b  = blockIdx.y;
  const int lensb = lens[b];
  const int bn   = lane & 15;
  const int acol = (lane >> 4) << 1;
  const int brow = (lane >> 4) << 1;
  const int rbase = (lane >> 4) << 3;
  float colmax = -INFINITY;
  for (int mt = 0; mt < KP / 16; ++mt) {
    v8f acc = {};
    const float* aptr = hrelu + ((size_t)b * KP + mt * 16 + bn) * HH + acol;
    for (int kk = 0; kk < HH; kk += 4) {
      v2f a = *(const v2f*)(aptr + kk);
      __builtin_prefetch(aptr + kk + 64, 0, 1);  // global_prefetch_b8
      const float* wp = w2 + (size_t)(kk + brow) * EE + n0 + bn;
      v2f bf;
      bf.x = wp[0]; bf.y = wp[EE];
      acc = wmma_f32_4(a, bf, acc);
    }
#pragma unroll
    for (int r = 0; r < 8; ++r) {
      int krow = mt * 16 + rbase + r;
      if (krow < lensb) colmax = fmaxf(colmax, acc[r]);
    }
  }
  // combine the two half-wave row groups holding the same columns
  colmax = fmaxf(colmax, __shfl_xor(colmax, 16, 32));
  if (lane < 16) pooled[(size_t)b * EE + n0 + lane] = colmax + b2[n0 + lane];
}

// ---------------------------------------------------------------------------
// Kernel 5: fused = pooled + global_features @ lin_w + lin_b.
// One wave per 16x16 tile of (B x E).
// ---------------------------------------------------------------------------
__global__ __launch_bounds__(256)
void k5_fuse(const float* __restrict__ gf, const float* __restrict__ lin_w,
             const float* __restrict__ lin_b, const float* __restrict__ pooled,
             float* __restrict__ fused) {
  const int wid  = threadIdx.x >> 5;
  const int lane = threadIdx.x & 31;
  const int gid  = blockIdx.x * 8 + wid;   // 0 .. 511
  const int m_t = gid >> 6;                // 0 .. 7
  const int n_t = gid & 63;                // 0 .. 63
  const int m0 = m_t * 16;
  const int n0 = n_t * 16;
  const int bn   = lane & 15;
  const int acol = (lane >> 4) << 1;
  const int brow = (lane >> 4) << 1;
  v8f acc = {};
  const float* aptr = gf + (size_t)(m0 + bn) * DD + acol;
  for (int kk = 0; kk < DD; kk += 4) {
    v2f a = *(const v2f*)(aptr + kk);
    const float* wp = lin_w + (size_t)(kk + brow) * EE + n0 + bn;
    v2f bf;
    bf.x = wp[0]; bf.y = wp[EE];
    acc = wmma_f32_4(a, bf, acc);
  }
  const int rbase = (lane >> 4) << 3;
  const int col = n0 + bn;
  const float lb = lin_b[col];
#pragma unroll
  for (int r = 0; r < 8; ++r) {
    const int row = m0 + rbase + r;
    fused[(size_t)row * EE + col] = acc[r] + lb + pooled[(size_t)row * EE + col];
  }
}

// ---------------------------------------------------------------------------
// Host launcher
// ---------------------------------------------------------------------------
extern "C" void kernel_launch(void* const* d_in, const int* in_sizes, int n_in,
                              void* d_out, int out_size, void* d_ws, size_t ws_size,
                              hipStream_t stream) {
  const float* gfeat = (const float*)d_in[0];
  // d_in[1] = features : unused by the reference
  const int*   text  = (const int*)d_in[2];
  const float* atten = (const float*)d_in[3];
  const float* lin_w = (const float*)d_in[4];
  const float* lin_b = (const float*)d_in[5];
  const float* w1    = (const float*)d_in[6];
  const float* b1    = (const float*)d_in[7];
  const float* g1    = (const float*)d_in[8];
  const float* be1   = (const float*)d_in[9];
  const float* rm1   = (const float*)d_in[10];
  const float* rv1   = (const float*)d_in[11];
  const float* w2    = (const float*)d_in[12];
  const float* b2    = (const float*)d_in[13];

  float* out    = (float*)d_out;
  float* fused  = out;                       // (B, E)
  float* pooled = out + (size_t)BB * EE;     // (B, E)

  // Workspace layout (bytes), 256-aligned sections
  char* ws = (char*)d_ws;
  int*   idxbuf = (int*)(ws + 0);            // B*K ints      = 78336 B
  int*   lensb  = (int*)(ws + 78336);        // B ints
  int*   eosb   = (int*)(ws + 78848);        // B ints
  float* gf     = (float*)(ws + 79360);      // B*D floats    = 262144 B
  float* localn = (float*)(ws + 341504);     // B*KP*D floats = 41943040 B
  float* hrelu  = (float*)(ws + 42284544);   // B*KP*H floats = 41943040 B

  k1_prep<<<BB, LL, 0, stream>>>(gfeat, text, atten, gf, idxbuf, lensb, eosb);
  k2_gather_norm<<<BB * KP, 256, 0, stream>>>(gfeat, idxbuf, localn);
  k3_gemm1<<<dim3(HH / BN, (BB * KP) / BM), 256, 0, stream>>>(
      localn, w1, b1, g1, be1, rm1, rv1, hrelu);
  k4_gemm2_pool<<<dim3(EE / 16, BB), 32, 0, stream>>>(hrelu, w2, b2, lensb, pooled);
  k5_fuse<<<(BB / 16) * (EE / 16) / 8, 256, 0, stream>>>(gf, lin_w, lin_b, pooled, fused);
}